// VectorQuantizer_29394756174026
// MI455X (gfx1250) — compile-verified
//
#include <hip/hip_runtime.h>
#include <hip/hip_bf16.h>
#include <math.h>

// Problem constants (from reference)
#define Bn   32
#define Cc   64
#define Hh   64
#define Ww   64
#define Kk   512
#define Dd   64
#define HW   (Hh * Ww)                 // 4096
#define CHW  (Cc * HW)                 // 262144
#define NTOT (Bn * HW)                 // 131072 flattened vectors

typedef float v2f __attribute__((ext_vector_type(2)));
typedef float v4f __attribute__((ext_vector_type(4)));
typedef float v8f __attribute__((ext_vector_type(8)));

// ---------------------------------------------------------------------------
// Prep: codebook row norms + zero the count histogram (deterministic each call)
// ---------------------------------------------------------------------------
__global__ __launch_bounds__(256) void vq_prep(const float* __restrict__ Wm,
                                               float* __restrict__ wn2,
                                               unsigned int* __restrict__ counts) {
    int k = blockIdx.x * 256 + threadIdx.x;
    if (k < Kk) {
        float s = 0.0f;
        #pragma unroll
        for (int d = 0; d < Dd; ++d) {
            float v = Wm[k * Dd + d];
            s += v * v;
        }
        wn2[k] = s;
        counts[k] = 0u;
    }
}

// ---------------------------------------------------------------------------
// Main: distances via V_WMMA_F32_16X16X4_F32, argmin, one-hot, quantized
// One block (256 threads = 8 waves) per 16-row tile of the N dimension.
// Wave w covers codebook columns [w*64, w*64+64) as 4 WMMA 16x16 tiles.
// Big output streams (distances, encodings) are stored non-temporal: they are
// write-once (536 MB total) and would otherwise flush the 192 MB L2 that keeps
// the codebook hot for all 8192 blocks.
// ---------------------------------------------------------------------------
__global__ __launch_bounds__(256) void vq_main(const float* __restrict__ x,
                                               const float* __restrict__ Wm,
                                               const float* __restrict__ wn2,
                                               float* __restrict__ out_dist,
                                               float* __restrict__ out_enc,
                                               float* __restrict__ out_idx,
                                               float* __restrict__ out_q,
                                               unsigned int* __restrict__ counts) {
    __shared__ float As[16 * 66];          // A tile, padded stride (bank-conflict free)
    __shared__ float xn[16];               // row norms of A
    __shared__ float cand_v[8][16][16];    // per-wave argmin candidates (value)
    __shared__ int   cand_i[8][16][16];    // per-wave argmin candidates (index)
    __shared__ int   idx_s[16];            // final argmin per row

    const int tid   = threadIdx.x;
    const int wave  = tid >> 5;
    const int lane  = tid & 31;
    const int lhalf = lane & 15;
    const int kb    = (lane >> 4) * 2;     // A/B k-sub-row base: lanes 0-15 -> K 0,1 ; 16-31 -> K 2,3
    const int rowbase = (lane >> 4) * 8;   // C/D rows this lane half holds (M = rowbase + vgpr)

    const int n0  = blockIdx.x * 16;       // first of 16 consecutive flat rows
    const int b   = n0 / HW;
    const int hw0 = n0 % HW;               // 16 consecutive n share (b,h); w contiguous
    const long xbase = (long)b * CHW + hw0;

    // ---- stage A[16 rows][64 ch] from NCHW (coalesced: 16 contiguous w per ch)
    #pragma unroll
    for (int j = 0; j < 4; ++j) {
        int e = tid + j * 256;
        int c = e >> 4;
        int m = e & 15;
        As[m * 66 + c] = x[xbase + (long)c * HW + m];
    }
    __syncthreads();

    if (tid < 16) {
        float s = 0.0f;
        #pragma unroll
        for (int d = 0; d < Dd; ++d) {
            float v = As[tid * 66 + d];
            s += v * v;
        }
        xn[tid] = s;
    }
    __syncthreads();

    // ---- preload A fragments for all 16 k-steps (reused across all 4 tiles)
    v2f afrag[16];
    #pragma unroll
    for (int s = 0; s < 16; ++s) {
        int d = 4 * s + kb;
        afrag[s][0] = As[lhalf * 66 + d];
        afrag[s][1] = As[lhalf * 66 + d + 1];
    }
    float xrow[8];
    #pragma unroll
    for (int r = 0; r < 8; ++r) xrow[r] = xn[rowbase + r];

    float bestv[8];
    int   besti[8];
    #pragma unroll
    for (int r = 0; r < 8; ++r) { bestv[r] = 3.4e38f; besti[r] = 0; }

    // ---- 4 column tiles of 16 codes each: 64 v_wmma_f32_16x16x4_f32 per wave
    for (int tt = 0; tt < 4; ++tt) {
        const int k0  = wave * 64 + tt * 16;
        const int col = k0 + lhalf;        // this lane's output column
        v8f acc = {0.f, 0.f, 0.f, 0.f, 0.f, 0.f, 0.f, 0.f};

        #pragma unroll
        for (int s = 0; s < 16; ++s) {
            int d = 4 * s + kb;
            v2f bfrag;
            bfrag[0] = Wm[col * Dd + d];
            bfrag[1] = Wm[col * Dd + d + 1];
            acc = __builtin_amdgcn_wmma_f32_16x16x4_f32(
                false, afrag[s], false, bfrag, (short)0, acc, false, false);
        }

        const float wn = wn2[col];
        #pragma unroll
        for (int r = 0; r < 8; ++r) {
            int   row  = rowbase + r;
            float dist = xrow[r] + wn - 2.0f * acc[r];
            __builtin_nontemporal_store(dist, &out_dist[(long)(n0 + row) * Kk + col]);
            if (dist < bestv[r]) { bestv[r] = dist; besti[r] = col; }
        }
    }

    // ---- cross-wave argmin reduction (first-index tie-break, as jnp.argmin)
    #pragma unroll
    for (int r = 0; r < 8; ++r) {
        cand_v[wave][rowbase + r][lhalf] = bestv[r];
        cand_i[wave][rowbase + r][lhalf] = besti[r];
    }
    __syncthreads();

    if (tid < 16) {
        float bv = 3.4e38f;
        int   bi = 0x7fffffff;
        for (int wv = 0; wv < 8; ++wv) {
            for (int l = 0; l < 16; ++l) {
                float v = cand_v[wv][tid][l];
                int   i = cand_i[wv][tid][l];
                if (v < bv || (v == bv && i < bi)) { bv = v; bi = i; }
            }
        }
        idx_s[tid] = bi;
        out_idx[n0 + tid] = (float)bi;
        atomicAdd(&counts[bi], 1u);
    }
    __syncthreads();

    // ---- one-hot encodings: 16 rows x 512, b128 non-temporal coalesced stores
    //      (rows are 512 floats -> a float4 never crosses a row boundary)
    for (int e4 = tid; e4 < (16 * Kk) / 4; e4 += 256) {
        int m  = e4 >> 7;                 // 128 float4 per row
        int k4 = (e4 & 127) * 4;
        int im = idx_s[m];
        v4f v;
        v[0] = (k4 + 0 == im) ? 1.0f : 0.0f;
        v[1] = (k4 + 1 == im) ? 1.0f : 0.0f;
        v[2] = (k4 + 2 == im) ? 1.0f : 0.0f;
        v[3] = (k4 + 3 == im) ? 1.0f : 0.0f;
        __builtin_nontemporal_store(v, (v4f*)&out_enc[(long)(n0 + m) * Kk + k4]);
    }

    // ---- quantized (straight-through == gathered codeword), back to NCHW
    #pragma unroll
    for (int j = 0; j < 4; ++j) {
        int e = tid + j * 256;
        int c = e >> 4;
        int m = e & 15;
        __builtin_nontemporal_store(Wm[idx_s[m] * Dd + c],
                                    &out_q[xbase + (long)c * HW + m]);
    }
}

// ---------------------------------------------------------------------------
// Perplexity: exp(-sum(p * log(p + 1e-10))), single block, deterministic tree
// ---------------------------------------------------------------------------
__global__ __launch_bounds__(512) void vq_perp(const unsigned int* __restrict__ counts,
                                               float* __restrict__ out_p) {
    __shared__ float red[512];
    int tid = threadIdx.x;
    float p = (float)counts[tid] / (float)NTOT;
    red[tid] = p * logf(p + 1e-10f);
    __syncthreads();
    for (int s = 256; s > 0; s >>= 1) {
        if (tid < s) red[tid] += red[tid + s];
        __syncthreads();
    }
    if (tid == 0) out_p[0] = expf(-red[0]);
}

// ---------------------------------------------------------------------------
extern "C" void kernel_launch(void* const* d_in, const int* in_sizes, int n_in,
                              void* d_out, int out_size, void* d_ws, size_t ws_size,
                              hipStream_t stream) {
    const float* x  = (const float*)d_in[0];   // [B,C,H,W] f32
    const float* Wm = (const float*)d_in[1];   // [K,D]     f32

    float* outf = (float*)d_out;
    const long NK = (long)NTOT * Kk;
    float* out_dist = outf;                         // [N,K]
    float* out_enc  = outf + NK;                    // [N,K]
    float* out_idx  = outf + 2 * NK;                // [N]
    float* out_q    = outf + 2 * NK + NTOT;         // [B,C,H,W]
    float* out_perp = outf + 2 * NK + NTOT + (long)Bn * CHW;  // [1]

    float*        wn2    = (float*)d_ws;            // K floats
    unsigned int* counts = (unsigned int*)((char*)d_ws + Kk * sizeof(float)); // K u32

    vq_prep<<<(Kk + 255) / 256, 256, 0, stream>>>(Wm, wn2, counts);
    vq_main<<<NTOT / 16, 256, 0, stream>>>(x, Wm, wn2, out_dist, out_enc,
                                           out_idx, out_q, counts);
    vq_perp<<<1, 512, 0, stream>>>(counts, out_perp);
}